// RAM_43301860278541
// MI455X (gfx1250) — compile-verified
//
#include <hip/hip_runtime.h>
#include <stdint.h>

typedef __attribute__((ext_vector_type(2))) float v2f;
typedef __attribute__((ext_vector_type(8))) float v8f;

#define N_S   16
#define C_IN  256
#define C_MID 64
#define H_IMG 128
#define W_IMG 128
#define HW    16384   // 128*128

// ---------------------------------------------------------------------------
// Kernel 1: faithful adaptive avg pool 128x128 -> 3x3 (overlapping bins)
// one block per (n,c) plane; LDS tree reduction per bin
// ---------------------------------------------------------------------------
__global__ void __launch_bounds__(256) pool_kernel(const float* __restrict__ x,
                                                   float* __restrict__ xp) {
    __shared__ float red[256];
    const int t = threadIdx.x;
    const int plane = blockIdx.x;                 // n*C_IN + c
    const float* px = x + (size_t)plane * HW;
    const int bs[3] = {0, 42, 85};                // floor(i*128/3)
    const int be[3] = {43, 86, 128};              // ceil((i+1)*128/3)

    for (int i = 0; i < 3; ++i) {
        for (int j = 0; j < 3; ++j) {
            const int sh = bs[i], eh = be[i], sw = bs[j], ew = be[j];
            const int ncols = ew - sw;
            const int cnt = (eh - sh) * ncols;
            float sum = 0.f;
            for (int e = t; e < cnt; e += 256) {
                const int r = sh + e / ncols;
                const int c = sw + e % ncols;
                sum += px[r * W_IMG + c];
            }
            __syncthreads();                      // protect red[] reuse
            red[t] = sum;
            __syncthreads();
            for (int st = 128; st > 0; st >>= 1) {
                if (t < st) red[t] += red[t + st];
                __syncthreads();
            }
            if (t == 0)
                xp[(size_t)plane * 9 + i * 3 + j] = red[0] / (float)cnt;
        }
    }
}

// ---------------------------------------------------------------------------
// Kernel 2: g[n][o][ij] = conv_w[o,:] @ xp[n,:,ij] + conv_b[o]   (tiny)
// ---------------------------------------------------------------------------
__global__ void __launch_bounds__(256) gfilter_kernel(const float* __restrict__ xp,
                                                      const float* __restrict__ conv_w,
                                                      const float* __restrict__ conv_b,
                                                      float* __restrict__ g) {
    const int t = blockIdx.x * 256 + threadIdx.x;   // t = ((n*64)+o)*9 + ij
    if (t >= N_S * C_MID * 9) return;
    const int n   = t / (C_MID * 9);
    const int rem = t % (C_MID * 9);
    const int o   = rem / 9;
    const int ij  = rem % 9;
    const float* w   = conv_w + (size_t)o * C_IN;
    const float* xpn = xp + (size_t)n * C_IN * 9 + ij;
    float sum = conv_b[o];
    for (int c = 0; c < C_IN; ++c) sum += w[c] * xpn[(size_t)c * 9];
    g[t] = sum;
}

// ---------------------------------------------------------------------------
// Kernel 3: GEMM1  f[n] = conv_w[64x256] @ x[n][256xHW] + conv_b
// wave tile: 64 rows (4 Mtiles) x 32 pixels (2 Ntiles), K-loop step 4 (WMMA f32)
// ---------------------------------------------------------------------------
__global__ void __launch_bounds__(256) conv1_wmma(const float* __restrict__ x,
                                                  const float* __restrict__ conv_w,
                                                  const float* __restrict__ conv_b,
                                                  float* __restrict__ f) {
    const int lane = threadIdx.x & 31;
    const int half = lane >> 4;          // 0: lanes 0-15, 1: lanes 16-31
    const int lm   = lane & 15;
    const int gw   = blockIdx.x * 8 + (threadIdx.x >> 5);  // global wave id
    const int s    = gw >> 9;            // sample (512 waves per sample)
    const int pb   = (gw & 511) * 32;    // pixel base
    const float* xs = x + (size_t)s * C_IN * HW;

    const v8f vzero = {0.f, 0.f, 0.f, 0.f, 0.f, 0.f, 0.f, 0.f};
    v8f acc[4][2];
#pragma unroll
    for (int mt = 0; mt < 4; ++mt)
#pragma unroll
        for (int nt = 0; nt < 2; ++nt) acc[mt][nt] = vzero;

    for (int k0 = 0; k0 < C_IN; k0 += 4) {
        const int ka = k0 + 2 * half;    // K pair handled by this half-wave
        v2f a[4];
#pragma unroll
        for (int mt = 0; mt < 4; ++mt) { // A: lane=M, vgpr/half=K (ISA layout)
            a[mt] = *(const v2f*)(conv_w + (size_t)(mt * 16 + lm) * C_IN + ka);
        }
        v2f b[2];
#pragma unroll
        for (int nt = 0; nt < 2; ++nt) { // B: lane=N, vgpr/half=K (mirror of A)
            const float* bp = xs + (size_t)ka * HW + pb + nt * 16 + lm;
            b[nt].x = bp[0];
            b[nt].y = bp[HW];
        }
#pragma unroll
        for (int mt = 0; mt < 4; ++mt)
#pragma unroll
            for (int nt = 0; nt < 2; ++nt)
                acc[mt][nt] = __builtin_amdgcn_wmma_f32_16x16x4_f32(
                    false, a[mt], false, b[nt], (short)0, acc[mt][nt], false, false);
    }

    // C/D layout: vgpr v holds M = v + 8*half, N = lane&15
#pragma unroll
    for (int mt = 0; mt < 4; ++mt) {
#pragma unroll
        for (int v = 0; v < 8; ++v) {
            const int ch = mt * 16 + v + 8 * half;
            const float bias = conv_b[ch];
#pragma unroll
            for (int nt = 0; nt < 2; ++nt)
                f[((size_t)s * C_MID + ch) * HW + pb + nt * 16 + lm] =
                    acc[mt][nt][v] + bias;
        }
    }
}

// ---------------------------------------------------------------------------
// Kernel 4: async-stage f halo tile -> LDS, dynamic depthwise 3x3 (+dw_b) in
// LDS, then GEMM2:  out[n] = fuse_w[256x64] @ y[64xHW] + fuse_b
// WG tile: 16(w) x 4(h) pixels of one sample; 8 waves x (2 Mtiles x 4 Ntiles)
// ---------------------------------------------------------------------------
#define FT_W 18            // halo tile width  (16 + 2)
#define FT_H 6             // halo tile height (4 + 2)
#define FT_PLANE (FT_W * FT_H)   // 108 floats per channel

__global__ void __launch_bounds__(256) dwfuse_wmma(const float* __restrict__ f,
                                                   const float* __restrict__ g,
                                                   const float* __restrict__ dw_b,
                                                   const float* __restrict__ fuse_w,
                                                   const float* __restrict__ fuse_b,
                                                   float* __restrict__ out) {
    __shared__ float ftile[C_MID * FT_PLANE];     // 27648 B  f halo (async-staged)
    __shared__ float ytile[C_MID * 64];           // 16384 B  [k=channel][n=pixel]
    __shared__ float gtile[C_MID * 9];            //  2304 B  per-sample dw filters
    const int b    = blockIdx.x;
    const int s    = b >> 8;                      // 256 tiles per sample
    const int tp   = b & 255;
    const int h0   = (tp >> 3) * 4;
    const int w0   = (tp & 7) * 16;
    const int tid  = threadIdx.x;

    // stage g[n] filters into LDS (tiny)
    for (int idx = tid; idx < C_MID * 9; idx += 256)
        gtile[idx] = g[(size_t)s * C_MID * 9 + idx];

    // async-stage f halo tile: rows h0-1..h0+4, cols w0-1..w0+16, all 64 ch.
    // In-image elements use GLOBAL_LOAD_ASYNC_TO_LDS_B32 (ASYNCcnt path);
    // out-of-image elements are zero-filled via normal ds stores (disjoint).
    for (int idx = tid; idx < C_MID * FT_PLANE; idx += 256) {
        const int c   = idx / FT_PLANE;
        const int rem = idx % FT_PLANE;
        const int r   = rem / FT_W;
        const int col = rem % FT_W;
        const int hh  = h0 - 1 + r;
        const int ww  = w0 - 1 + col;
        if (hh >= 0 && hh < H_IMG && ww >= 0 && ww < W_IMG) {
            const float* gp = f + ((size_t)s * C_MID + c) * HW + hh * W_IMG + ww;
            // flat-pointer low 32 bits == LDS byte offset (aperture truncation)
            const unsigned lds_off = (unsigned)(uintptr_t)(&ftile[idx]);
            asm volatile("global_load_async_to_lds_b32 %0, %1, off"
                         :: "v"(lds_off),
                            "v"((unsigned long long)(uintptr_t)gp)
                         : "memory");
        } else {
            ftile[idx] = 0.f;
        }
    }
    asm volatile("s_wait_asynccnt 0x0" ::: "memory");
    __syncthreads();

    // build y tile from LDS: y = depthwise3x3(f, g) + dw_b
    for (int idx = tid; idx < C_MID * 64; idx += 256) {
        const int c  = idx >> 6;
        const int p  = idx & 63;
        const int pr = p >> 4;
        const int pw = p & 15;
        const float* fc = &ftile[c * FT_PLANE];
        const float* gc = &gtile[c * 9];
        float sum = dw_b[c];
#pragma unroll
        for (int dy = 0; dy < 3; ++dy)
#pragma unroll
            for (int dx = 0; dx < 3; ++dx)
                sum += gc[dy * 3 + dx] * fc[(pr + dy) * FT_W + (pw + dx)];
        ytile[c * 64 + p] = sum;
    }
    __syncthreads();

    const int lane = tid & 31;
    const int half = lane >> 4;
    const int lm   = lane & 15;
    const int wv   = tid >> 5;                    // wave 0..7 -> Mtiles 2wv..2wv+1

    const v8f vzero = {0.f, 0.f, 0.f, 0.f, 0.f, 0.f, 0.f, 0.f};
    v8f acc[2][4];
#pragma unroll
    for (int im = 0; im < 2; ++im)
#pragma unroll
        for (int j = 0; j < 4; ++j) acc[im][j] = vzero;

    for (int kk = 0; kk < 16; ++kk) {             // K = 64, step 4
        const int ka = kk * 4 + 2 * half;
        v2f a[2];
#pragma unroll
        for (int im = 0; im < 2; ++im) {
            const int row = (wv * 2 + im) * 16 + lm;
            a[im] = *(const v2f*)(fuse_w + (size_t)row * C_MID + ka);
        }
        v2f bf[4];
#pragma unroll
        for (int j = 0; j < 4; ++j) {
            const int n = j * 16 + lm;
            bf[j].x = ytile[ka * 64 + n];
            bf[j].y = ytile[(ka + 1) * 64 + n];
        }
#pragma unroll
        for (int im = 0; im < 2; ++im)
#pragma unroll
            for (int j = 0; j < 4; ++j)
                acc[im][j] = __builtin_amdgcn_wmma_f32_16x16x4_f32(
                    false, a[im], false, bf[j], (short)0, acc[im][j], false, false);
    }

    // Ntile j == tile row j -> coalesced row-segment stores
#pragma unroll
    for (int im = 0; im < 2; ++im) {
#pragma unroll
        for (int v = 0; v < 8; ++v) {
            const int ch = (wv * 2 + im) * 16 + v + 8 * half;
            const float bias = fuse_b[ch];
#pragma unroll
            for (int j = 0; j < 4; ++j) {
                const int h = h0 + j;
                out[((size_t)s * C_IN + ch) * HW + h * W_IMG + w0 + lm] =
                    acc[im][j][v] + bias;
            }
        }
    }
}

// ---------------------------------------------------------------------------
extern "C" void kernel_launch(void* const* d_in, const int* in_sizes, int n_in,
                              void* d_out, int out_size, void* d_ws, size_t ws_size,
                              hipStream_t stream) {
    (void)in_sizes; (void)n_in; (void)out_size; (void)ws_size;
    const float* x      = (const float*)d_in[0];   // [16,256,128,128]
    const float* conv_w = (const float*)d_in[1];   // [64,256]
    const float* conv_b = (const float*)d_in[2];   // [64]
    const float* dw_b   = (const float*)d_in[3];   // [64]
    const float* fuse_w = (const float*)d_in[4];   // [256,64]
    const float* fuse_b = (const float*)d_in[5];   // [256]
    float* out = (float*)d_out;                    // [16,256,128,128]

    // workspace layout: f (16.78M floats) | xp (36864) | g (9216)   ~67.3 MB
    float* fbuf = (float*)d_ws;
    float* xp   = fbuf + (size_t)N_S * C_MID * HW;
    float* gbuf = xp + (size_t)N_S * C_IN * 9;

    pool_kernel<<<N_S * C_IN, 256, 0, stream>>>(x, xp);
    gfilter_kernel<<<(N_S * C_MID * 9 + 255) / 256, 256, 0, stream>>>(xp, conv_w, conv_b, gbuf);
    conv1_wmma<<<(N_S * (HW / 32)) / 8, 256, 0, stream>>>(x, conv_w, conv_b, fbuf);
    dwfuse_wmma<<<N_S * 256, 256, 0, stream>>>(fbuf, gbuf, dw_b, fuse_w, fuse_b, out);
}